// FinanceMoEModel_53618371723862
// MI455X (gfx1250) — compile-verified
//
#include <hip/hip_runtime.h>

typedef __attribute__((ext_vector_type(2))) float v2f;
typedef __attribute__((ext_vector_type(8))) float v8f;

__device__ __forceinline__ unsigned rotl32(unsigned x, unsigned r) {
  return (x << r) | (x >> (32u - r));
}

// Faithful JAX threefry2x32 normal for key(42), shape (B*S,), element idx.
__device__ __forceinline__ float jax_threefry_normal(unsigned idx, unsigned nhalf) {
  unsigned c0, c1; bool hi;
  if (idx < nhalf) { c0 = idx;          c1 = idx + nhalf; hi = false; }
  else             { c0 = idx - nhalf;  c1 = idx;         hi = true;  }
  const unsigned k0 = 0u, k1 = 42u;
  const unsigned k2 = k0 ^ k1 ^ 0x1BD11BDAu;
  unsigned x0 = c0 + k0, x1 = c1 + k1;
#define TF_ROUND(r) { x0 += x1; x1 = rotl32(x1, r); x1 ^= x0; }
  TF_ROUND(13) TF_ROUND(15) TF_ROUND(26) TF_ROUND(6)
  x0 += k1; x1 += k2 + 1u;
  TF_ROUND(17) TF_ROUND(29) TF_ROUND(16) TF_ROUND(24)
  x0 += k2; x1 += k0 + 2u;
  TF_ROUND(13) TF_ROUND(15) TF_ROUND(26) TF_ROUND(6)
  x0 += k0; x1 += k1 + 3u;
  TF_ROUND(17) TF_ROUND(29) TF_ROUND(16) TF_ROUND(24)
  x0 += k1; x1 += k2 + 4u;
  TF_ROUND(13) TF_ROUND(15) TF_ROUND(26) TF_ROUND(6)
  x0 += k2; x1 += k0 + 5u;
#undef TF_ROUND
  unsigned bits = hi ? x1 : x0;
  float f = __uint_as_float((bits >> 9) | 0x3f800000u) - 1.0f;  // [0,1)
  const float lo = -0.99999994f;                                 // nextafter(-1,0)
  float u = fmaxf(lo, f * (1.0f - lo) + lo);
  return 1.4142135623730951f * erfinvf(u);
}

__device__ __forceinline__ float sigmoidf(float x) {
  return 1.0f / (1.0f + expf(-x));
}

// One wave handles 16 sequence positions per iteration.
// WMMA: D(16x16) = W_pad(16x16, M=expert, K=H) x X^T(16x16, K=H, N=pos) + bias
// -> lane l (<16): acc[0..5] = logits for position (tile*16 + l), experts 0..5.
__global__ __launch_bounds__(256)
void finance_moe_fwd_kernel(const float* __restrict__ X,   // (B*S, 16)
                            const float* __restrict__ W,   // (6, 16)
                            const float* __restrict__ Bv,  // (6,)
                            float* __restrict__ pred,      // (B*S)
                            float* __restrict__ rout,      // (B*S, 6)
                            float* __restrict__ asg,       // (B*S)
                            int nPos, int nTiles) {
  const int lane   = threadIdx.x & 31;
  const int laneHi = lane >> 4;   // 0: K lanes {4j,4j+1}, 1: K lanes {4j+2,4j+3}
  const int r      = lane & 15;   // M row for A (expert), N col for B (position)
  const int waveId = (blockIdx.x * blockDim.x + threadIdx.x) >> 5;
  const int nWaves = (gridDim.x * blockDim.x) >> 5;

  // Loop-invariant A-matrix (router weights, rows >=6 zero-padded).
  // A layout per WMMA j: lane holds W_pad[r][4j + 2*laneHi + {0,1}].
  v2f a0, a1, a2, a3;
  {
    const bool valid = (r < 6);
    const int base = r * 16 + 2 * laneHi;
    a0[0] = valid ? W[base + 0]  : 0.0f;  a0[1] = valid ? W[base + 1]  : 0.0f;
    a1[0] = valid ? W[base + 4]  : 0.0f;  a1[1] = valid ? W[base + 5]  : 0.0f;
    a2[0] = valid ? W[base + 8]  : 0.0f;  a2[1] = valid ? W[base + 9]  : 0.0f;
    a3[0] = valid ? W[base + 12] : 0.0f;  a3[1] = valid ? W[base + 13] : 0.0f;
  }
  // C init = bias: C vgpr g, lanes<16 -> row M=g (expert g), lanes>=16 -> row g+8 (pad).
  v8f cinit;
#pragma unroll
  for (int g = 0; g < 8; ++g)
    cinit[g] = (laneHi == 0 && g < 6) ? Bv[g] : 0.0f;

  for (int t = waveId; t < nTiles; t += nWaves) {
    const long p0 = (long)t << 4;
    const float* xb = X + ((p0 + r) << 4) + 2 * laneHi;
    // B layout per WMMA j: lane holds X^T[4j + 2*laneHi + {0,1}][r] = x[p0+r][k..k+1]
    v2f b0 = *(const v2f*)(xb + 0);
    v2f b1 = *(const v2f*)(xb + 4);
    v2f b2 = *(const v2f*)(xb + 8);
    v2f b3 = *(const v2f*)(xb + 12);

    if (t + nWaves < nTiles)
      __builtin_prefetch(xb + (long)nWaves * 256, 0, 3);

    // Router GEMM on the matrix pipe: 4x K=4 accumulation.
    v8f acc = cinit;
    acc = __builtin_amdgcn_wmma_f32_16x16x4_f32(false, a0, false, b0, (short)0, acc, false, false);
    acc = __builtin_amdgcn_wmma_f32_16x16x4_f32(false, a1, false, b1, (short)0, acc, false, false);
    acc = __builtin_amdgcn_wmma_f32_16x16x4_f32(false, a2, false, b2, (short)0, acc, false, false);
    acc = __builtin_amdgcn_wmma_f32_16x16x4_f32(false, a3, false, b3, (short)0, acc, false, false);

    // Per-lane partial stats over this lane's 8 K-values.
    // laneHi==0 owns k = {0,1,4,5,8,9,12,13}; laneHi==1 owns k = {2,3,6,7,10,11,14,15}.
    const float b0s = b0[0] + b0[1], b1s = b1[0] + b1[1];
    const float b2s = b2[0] + b2[1], b3s = b3[0] + b3[1];
    float sall = b0s + b1s + b2s + b3s;
    float ssq  = b0[0]*b0[0] + b0[1]*b0[1] + b1[0]*b1[0] + b1[1]*b1[1]
               + b2[0]*b2[0] + b2[1]*b2[1] + b3[0]*b3[0] + b3[1]*b3[1];
    float s4   = b0s;                              // k<4
    float s6   = laneHi ? b0s : (b0s + b1s);       // k<6
    float s8   = b0s + b1s;                        // k<8
    float s610 = laneHi ? b1s : b2s;               // 6<=k<10

    // Combine complementary halves (lane l <-> l+16).
    sall += __shfl_xor(sall, 16, 32);
    ssq  += __shfl_xor(ssq,  16, 32);
    s4   += __shfl_xor(s4,   16, 32);
    s6   += __shfl_xor(s6,   16, 32);
    s8   += __shfl_xor(s8,   16, 32);
    s610 += __shfl_xor(s610, 16, 32);

    if (laneHi == 0) {
      const long p = p0 + r;
      float l0 = acc[0], l1 = acc[1], l2 = acc[2],
            l3 = acc[3], l4 = acc[4], l5 = acc[5];

      // argmax(softmax(logits)) == first-occurrence argmax(logits)
      float mx = l0; int am = 0;
      if (l1 > mx) { mx = l1; am = 1; }
      if (l2 > mx) { mx = l2; am = 2; }
      if (l3 > mx) { mx = l3; am = 3; }
      if (l4 > mx) { mx = l4; am = 4; }
      if (l5 > mx) { mx = l5; am = 5; }

      const float e0 = expf(l0 - mx), e1 = expf(l1 - mx), e2 = expf(l2 - mx);
      const float e3 = expf(l3 - mx), e4 = expf(l4 - mx), e5 = expf(l5 - mx);
      const float inv = 1.0f / (e0 + e1 + e2 + e3 + e4 + e5);

      const float mean = sall * 0.0625f;
      const float var  = (ssq - sall * sall * 0.0625f) * (1.0f / 15.0f);
      const float sd   = sqrtf(fmaxf(var, 0.0f));

      float prv;
      switch (am) {
        case 0: prv = tanhf(s4 * 0.25f) * (1.0f + sd); break;
        case 1: prv = sigmoidf(mean) * 0.3f - 0.15f; break;
        case 2: prv = (s6 * (1.0f / 6.0f)) * 0.8f
                    + sinf((s610 * 0.25f) * 3.14159f) * 0.4f; break;
        case 3: prv = tanhf(s8 * 0.125f) * 0.9f
                    + 0.05f * jax_threefry_normal((unsigned)p, (unsigned)(nPos >> 1)); break;
        case 4: prv = powf(fmaxf(mean, 0.0f), 1.2f) + sd * 2.5f - 0.5f; break;
        default: prv = sigmoidf(mean) * 0.4f + tanhf(sd) * 0.2f; break;
      }

      pred[p] = prv;
      float* rp = rout + p * 6;                 // 24B stride -> 8B aligned
      v2f r01 = { e0 * inv, e1 * inv };
      v2f r23 = { e2 * inv, e3 * inv };
      v2f r45 = { e4 * inv, e5 * inv };
      *(v2f*)(rp + 0) = r01;
      *(v2f*)(rp + 2) = r23;
      *(v2f*)(rp + 4) = r45;
      asg[p] = (float)am;
    }
  }
}

extern "C" void kernel_launch(void* const* d_in, const int* in_sizes, int n_in,
                              void* d_out, int out_size, void* d_ws, size_t ws_size,
                              hipStream_t stream) {
  const float* X  = (const float*)d_in[0];   // sequence_embeddings (B,S,16)
  // d_in[1] market_volatility, d_in[2] risk_factors: forward-dead
  // (straight-through: routing_probs value == probs_base exactly).
  const float* W  = (const float*)d_in[3];   // router_w (6,16)
  const float* Bv = (const float*)d_in[4];   // router_b (6,)

  const int nPos   = in_sizes[1];            // B*S
  const int nTiles = nPos >> 4;              // 16 positions per wave-tile

  float* out  = (float*)d_out;
  float* pred = out;                                   // (B*S)
  float* rout = out + (size_t)nPos;                    // (B*S,6)
  float* asg  = out + (size_t)nPos * 7;                // (B*S)

  int blocks = (nTiles * 32 + 255) / 256;    // 1 tile per wave if small
  if (blocks > 1024) blocks = 1024;          // else persistent waves, ~8 tiles each
  if (blocks < 1) blocks = 1;

  finance_moe_fwd_kernel<<<dim3(blocks), dim3(256), 0, stream>>>(
      X, W, Bv, pred, rout, asg, nPos, nTiles);
}